// FasterRCNN_39470749450617
// MI455X (gfx1250) — compile-verified
//
#include <hip/hip_runtime.h>
#include <cmath>

// ---------------- problem constants ----------------
#define IMG_H   38
#define IMG_W   63
#define SPAT    (IMG_H*IMG_W)      // 2394
#define CIN     512
#define CONVK   (CIN*9)            // 4608
#define NANCH   9
#define NTOT    (SPAT*NANCH)       // 21546
#define PRE_NMS 6000
#define POST_NMS 300
#define POOL    7
#define FCK     (CIN*POOL*POOL)    // 25088
#define MPAD    304                // 19 * 16
#define MT_FC   19
#define NCLS    32

typedef __attribute__((ext_vector_type(2))) float v2f;
typedef __attribute__((ext_vector_type(8))) float v8f;

// ---------------- workspace layout (floats) ----------------
#define OFF_RPN    ((size_t)0)                         // 2394*512
#define OFF_CLS    (OFF_RPN  + (size_t)SPAT*CIN)       // 2394*18
#define OFF_BBX    (OFF_CLS  + (size_t)SPAT*18)        // 2394*36
#define OFF_PROPS  (OFF_BBX  + (size_t)SPAT*36)        // 21546*4
#define OFF_SCORE  (OFF_PROPS+ (size_t)NTOT*4)         // 21546
#define OFF_SCNMS  (OFF_SCORE+ (size_t)NTOT)           // 21546
#define OFF_ROIS   (OFF_SCNMS+ (size_t)NTOT)           // 300*4
#define OFF_XP     (OFF_ROIS + (size_t)POST_NMS*4)     // 304*25088
#define OFF_X2     (OFF_XP   + (size_t)MPAD*FCK)       // 304*4096
#define OFF_X3     (OFF_X2   + (size_t)MPAD*4096)      // 304*4096
#define OFF_CRAW   (OFF_X3   + (size_t)MPAD*4096)      // 304*32

// =====================================================================
// K1: RPN 3x3 conv (512->512, pad 1) + bias + ReLU, implicit-GEMM WMMA f32
//     M = spatial (2394), N = out channel (512), K = 4608
//     output layout: rpn[s*512 + c]
// =====================================================================
__global__ __launch_bounds__(32) void rpn_conv_wmma(
    const float* __restrict__ img, const float* __restrict__ w,
    const float* __restrict__ bias, float* __restrict__ out)
{
  const int nt = blockIdx.x;           // 0..31  (N tiles)
  const int mt = blockIdx.y;           // 0..149 (M tiles)
  const int lane = threadIdx.x;
  const int ln = lane & 15, hi = lane >> 4;
  const int n0 = nt * 16, m0 = mt * 16;

  const int srow = m0 + ln;            // spatial row this lane supplies for A
  const int h = srow / IMG_W, wx = srow - h * IMG_W;
  const bool arow_ok = (srow < SPAT);

  v8f acc = {};
  const float* wb = w + (size_t)(n0 + ln) * CONVK;

  for (int k = 0; k < CONVK; k += 4) {
    const int ka = k + 2 * hi;
    v2f bf; bf.x = wb[ka]; bf.y = wb[ka + 1];
    v2f af;
    #pragma unroll
    for (int j = 0; j < 2; ++j) {
      const int kk = ka + j;
      const int ci = kk / 9;
      const int r9 = kk - ci * 9;
      const int kr = r9 / 3, kc = r9 - kr * 3;
      const int ih = h + kr - 1, iw = wx + kc - 1;
      float v = 0.f;
      if (arow_ok && ih >= 0 && ih < IMG_H && iw >= 0 && iw < IMG_W)
        v = img[(size_t)ci * SPAT + ih * IMG_W + iw];
      af[j] = v;
    }
    acc = __builtin_amdgcn_wmma_f32_16x16x4_f32(false, af, false, bf,
                                                (short)0, acc, false, false);
  }
  const float bv = bias[n0 + ln];
  #pragma unroll
  for (int v = 0; v < 8; ++v) {
    const int s = m0 + v + 8 * hi;
    if (s < SPAT) {
      float o = acc[v] + bv;
      out[(size_t)s * CIN + n0 + ln] = fmaxf(o, 0.f);
    }
  }
}

// =====================================================================
// K2: 1x1 conv heads: cls_score (18ch) + bbox_pred (36ch)
// =====================================================================
__global__ void rpn_heads(const float* __restrict__ rpn,
                          const float* __restrict__ sw, const float* __restrict__ sb,
                          const float* __restrict__ bw, const float* __restrict__ bb,
                          float* __restrict__ cls_s, float* __restrict__ bbox_p)
{
  const int idx = blockIdx.x * blockDim.x + threadIdx.x;
  if (idx >= SPAT * 54) return;
  const int s = idx / 54, c = idx - s * 54;
  const float* x = rpn + (size_t)s * CIN;
  const float* wr; float acc;
  if (c < 18) { wr = sw + (size_t)c * CIN;        acc = sb[c]; }
  else        { wr = bw + (size_t)(c - 18) * CIN; acc = bb[c - 18]; }
  #pragma unroll 4
  for (int k = 0; k < CIN; ++k) acc += x[k] * wr[k];
  if (c < 18) cls_s[(size_t)s * 18 + c]        = acc;
  else        bbox_p[(size_t)s * 36 + (c - 18)] = acc;
}

// =====================================================================
// K3: proposals: sigmoid fg score, bbox_transform_inv, clip, min-size
// =====================================================================
struct AnchorTab { float a[36]; };

__global__ void proposals_kernel(const float* __restrict__ cls_s,
                                 const float* __restrict__ bbox_p,
                                 const float* __restrict__ info,
                                 AnchorTab at,
                                 float* __restrict__ props,
                                 float* __restrict__ scores)
{
  const int idx = blockIdx.x * blockDim.x + threadIdx.x;
  if (idx >= NTOT) return;
  const int s = idx / NANCH, a = idx - s * NANCH;
  const int h = s / IMG_W, w = s - h * IMG_W;

  // fg prob = softmax over {bg,fg} logits picking fg = sigmoid(s1 - s0)
  const float s0 = cls_s[(size_t)s * 18 + a];
  const float s1 = cls_s[(size_t)s * 18 + NANCH + a];
  const float score = 1.f / (1.f + expf(s0 - s1));

  const float ax1 = (float)(w * 16) + at.a[a * 4 + 0];
  const float ay1 = (float)(h * 16) + at.a[a * 4 + 1];
  const float ax2 = (float)(w * 16) + at.a[a * 4 + 2];
  const float ay2 = (float)(h * 16) + at.a[a * 4 + 3];
  const float aw = ax2 - ax1 + 1.f, ah = ay2 - ay1 + 1.f;
  const float cx = ax1 + 0.5f * aw,  cy = ay1 + 0.5f * ah;

  const float dx = bbox_p[(size_t)s * 36 + a * 4 + 0];
  const float dy = bbox_p[(size_t)s * 36 + a * 4 + 1];
  const float dw = bbox_p[(size_t)s * 36 + a * 4 + 2];
  const float dh = bbox_p[(size_t)s * 36 + a * 4 + 3];

  const float pcx = dx * aw + cx, pcy = dy * ah + cy;
  const float pw  = expf(dw) * aw, ph = expf(dh) * ah;

  const float im_h = info[0], im_w = info[1], im_sc = info[2];
  float x1 = fminf(fmaxf(pcx - 0.5f * pw, 0.f), im_w - 1.f);
  float y1 = fminf(fmaxf(pcy - 0.5f * ph, 0.f), im_h - 1.f);
  float x2 = fminf(fmaxf(pcx + 0.5f * pw, 0.f), im_w - 1.f);
  float y2 = fminf(fmaxf(pcy + 0.5f * ph, 0.f), im_h - 1.f);

  props[(size_t)idx * 4 + 0] = x1;
  props[(size_t)idx * 4 + 1] = y1;
  props[(size_t)idx * 4 + 2] = x2;
  props[(size_t)idx * 4 + 3] = y2;

  const bool ok = (x2 - x1 + 1.f >= 16.f * im_sc) && (y2 - y1 + 1.f >= 16.f * im_sc);
  scores[idx] = ok ? score : -__builtin_inff();
}

// =====================================================================
// K4: top-PRE_NMS mask via O(N^2) rank count (LDS-tiled)
// =====================================================================
__global__ __launch_bounds__(256) void topk_mask(const float* __restrict__ scores,
                                                 float* __restrict__ out_sc)
{
  __shared__ float tile[256];
  const int i = blockIdx.x * 256 + threadIdx.x;
  const float si = (i < NTOT) ? scores[i] : -__builtin_inff();
  int rank = 0;
  for (int t = 0; t < NTOT; t += 256) {
    const int j = t + threadIdx.x;
    tile[threadIdx.x] = (j < NTOT) ? scores[j] : -__builtin_inff();
    __syncthreads();
    const int lim = (NTOT - t < 256) ? (NTOT - t) : 256;
    for (int u = 0; u < lim; ++u) {
      const float sj = tile[u];
      const int j2 = t + u;
      if (sj > si || (sj == si && j2 < i)) ++rank;
    }
    __syncthreads();
  }
  if (i < NTOT) out_sc[i] = (rank < PRE_NMS) ? si : -__builtin_inff();
}

// =====================================================================
// K5: sequential NMS, single workgroup, 300 iterations of argmax+suppress
// =====================================================================
__global__ __launch_bounds__(1024) void nms_kernel(const float* __restrict__ props,
                                                   float* __restrict__ sc,
                                                   float* __restrict__ rois_ws,
                                                   float* __restrict__ rois_out)
{
  __shared__ float bv[1024];
  __shared__ int   bi[1024];
  __shared__ float box[4];
  __shared__ float barea;
  __shared__ int   gbest;
  const int tid = threadIdx.x;

  for (int it = 0; it < POST_NMS; ++it) {
    float best = -__builtin_inff(); int bidx = NTOT;
    for (int i = tid; i < NTOT; i += 1024) {
      const float v = sc[i];
      if (v > best || (v == best && i < bidx)) { best = v; bidx = i; }
    }
    bv[tid] = best; bi[tid] = bidx;
    __syncthreads();
    for (int off = 512; off > 0; off >>= 1) {
      if (tid < off) {
        if (bv[tid + off] > bv[tid] ||
            (bv[tid + off] == bv[tid] && bi[tid + off] < bi[tid])) {
          bv[tid] = bv[tid + off]; bi[tid] = bi[tid + off];
        }
      }
      __syncthreads();
    }
    if (tid == 0) {
      int i = bi[0];
      if (it == 0) gbest = i;
      if (bv[0] == -__builtin_inff()) i = gbest;    // degenerate fallback
      const float x1 = props[(size_t)i * 4 + 0], y1 = props[(size_t)i * 4 + 1];
      const float x2 = props[(size_t)i * 4 + 2], y2 = props[(size_t)i * 4 + 3];
      box[0] = x1; box[1] = y1; box[2] = x2; box[3] = y2;
      barea = (x2 - x1 + 1.f) * (y2 - y1 + 1.f);
      rois_ws[it * 4 + 0] = x1; rois_ws[it * 4 + 1] = y1;
      rois_ws[it * 4 + 2] = x2; rois_ws[it * 4 + 3] = y2;
      rois_out[it * 4 + 0] = x1; rois_out[it * 4 + 1] = y1;
      rois_out[it * 4 + 2] = x2; rois_out[it * 4 + 3] = y2;
    }
    __syncthreads();
    const float x1 = box[0], y1 = box[1], x2 = box[2], y2 = box[3], ba = barea;
    for (int i = tid; i < NTOT; i += 1024) {
      const float bx1 = props[(size_t)i * 4 + 0], by1 = props[(size_t)i * 4 + 1];
      const float bx2 = props[(size_t)i * 4 + 2], by2 = props[(size_t)i * 4 + 3];
      const float xx1 = fmaxf(x1, bx1), yy1 = fmaxf(y1, by1);
      const float xx2 = fminf(x2, bx2), yy2 = fminf(y2, by2);
      const float inter = fmaxf(xx2 - xx1 + 1.f, 0.f) * fmaxf(yy2 - yy1 + 1.f, 0.f);
      const float area = (bx2 - bx1 + 1.f) * (by2 - by1 + 1.f);
      const float iou = inter / (ba + area - inter);
      if (iou > 0.7f) sc[i] = -__builtin_inff();
    }
    __syncthreads();
  }
}

// =====================================================================
// K6: ROI max-pool (on img_features) -> XP[304 x 25088] (c*49+ph*7+pw)
// =====================================================================
__global__ void roi_pool_kernel(const float* __restrict__ feat,
                                const float* __restrict__ rois,
                                float* __restrict__ xp)
{
  int idx = blockIdx.x * blockDim.x + threadIdx.x;
  if (idx >= POST_NMS * CIN * POOL * POOL) return;
  const int pw = idx % POOL; idx /= POOL;
  const int ph = idx % POOL; idx /= POOL;
  const int c  = idx % CIN;
  const int r  = idx / CIN;

  const float* roi = rois + (size_t)r * 4;
  const float rsw = rintf(roi[0] * 0.0625f), rsh = rintf(roi[1] * 0.0625f);
  const float rew = rintf(roi[2] * 0.0625f), reh = rintf(roi[3] * 0.0625f);
  const float bw = fmaxf(rew - rsw + 1.f, 1.f) / (float)POOL;
  const float bh = fmaxf(reh - rsh + 1.f, 1.f) / (float)POOL;

  const int h0 = (int)fminf(fmaxf(floorf(ph * bh) + rsh, 0.f), (float)IMG_H);
  const int h1 = (int)fminf(fmaxf(ceilf((ph + 1) * bh) + rsh, 0.f), (float)IMG_H);
  const int w0 = (int)fminf(fmaxf(floorf(pw * bw) + rsw, 0.f), (float)IMG_W);
  const int w1 = (int)fminf(fmaxf(ceilf((pw + 1) * bw) + rsw, 0.f), (float)IMG_W);

  float m = -1e30f;
  for (int hh = h0; hh < h1; ++hh)
    for (int ww = w0; ww < w1; ++ww)
      m = fmaxf(m, feat[(size_t)c * SPAT + hh * IMG_W + ww]);
  if (m < -1e29f) m = 0.f;
  xp[(size_t)r * FCK + c * (POOL * POOL) + ph * POOL + pw] = m;
}

// =====================================================================
// K7: generic FC GEMM with WMMA f32 16x16x4.
//     out[M x N] = act( A[MPAD x K] @ W[N x K]^T + bias )
//     One wave per 16-col N tile; all MT=19 M-tile accumulators resident
//     => each weight element streams from HBM exactly once; the 30 MB
//     activation matrix stays L2-resident across N-tile workgroups.
// =====================================================================
template<int MT>
__global__ __launch_bounds__(32) void fc_gemm_wmma(
    const float* __restrict__ A, const float* __restrict__ W,
    const float* __restrict__ bias, float* __restrict__ out,
    int K, int ldo, int Mvalid, int relu)
{
  const int n0 = blockIdx.x * 16;
  const int lane = threadIdx.x, ln = lane & 15, hi = lane >> 4;

  v8f acc[MT];
  #pragma unroll
  for (int m = 0; m < MT; ++m) acc[m] = (v8f){};

  const float* wr = W + (size_t)(n0 + ln) * K;
  for (int k = 0; k < K; k += 4) {
    const int ka = k + 2 * hi;
    v2f bf; bf.x = wr[ka]; bf.y = wr[ka + 1];
    #pragma unroll
    for (int m = 0; m < MT; ++m) {
      const float* ar = A + (size_t)(m * 16 + ln) * K + ka;
      v2f af; af.x = ar[0]; af.y = ar[1];
      acc[m] = __builtin_amdgcn_wmma_f32_16x16x4_f32(false, af, false, bf,
                                                     (short)0, acc[m], false, false);
    }
  }
  const float bv = bias[n0 + ln];
  #pragma unroll
  for (int m = 0; m < MT; ++m) {
    #pragma unroll
    for (int v = 0; v < 8; ++v) {
      const int row = m * 16 + v + 8 * hi;
      if (row < Mvalid) {
        float o = acc[m][v] + bv;
        if (relu) o = fmaxf(o, 0.f);
        out[(size_t)row * ldo + n0 + ln] = o;
      }
    }
  }
}

// =====================================================================
// K8: final 32-class softmax
// =====================================================================
__global__ void softmax32(const float* __restrict__ cls, float* __restrict__ out)
{
  const int r = blockIdx.x * blockDim.x + threadIdx.x;
  if (r >= POST_NMS) return;
  const float* x = cls + (size_t)r * NCLS;
  float m = x[0];
  for (int c = 1; c < NCLS; ++c) m = fmaxf(m, x[c]);
  float ssum = 0.f; float e[NCLS];
  for (int c = 0; c < NCLS; ++c) { e[c] = expf(x[c] - m); ssum += e[c]; }
  const float inv = 1.f / ssum;
  for (int c = 0; c < NCLS; ++c) out[(size_t)r * NCLS + c] = e[c] * inv;
}

__global__ void zero_fill(float* __restrict__ p, size_t n)
{
  const size_t i = (size_t)blockIdx.x * blockDim.x + threadIdx.x;
  if (i < n) p[i] = 0.f;
}

// =====================================================================
// host: anchor table (round-half-even matches np.round)
// =====================================================================
static void gen_anchors(float* out36)
{
  const double xc = 7.5, yc = 7.5, size = 256.0;
  const double ratios[3] = {0.5, 1.0, 2.0};
  const double scales[3] = {8.0, 16.0, 32.0};
  for (int ri = 0; ri < 3; ++ri) {
    const double ws = nearbyint(sqrt(size / ratios[ri]));
    const double hs = nearbyint(ws * ratios[ri]);
    const double ax1 = xc - 0.5 * (ws - 1), ay1 = yc - 0.5 * (hs - 1);
    const double ax2 = xc + 0.5 * (ws - 1), ay2 = yc + 0.5 * (hs - 1);
    const double aw = ax2 - ax1 + 1, ah = ay2 - ay1 + 1;
    const double acx = ax1 + 0.5 * (aw - 1), acy = ay1 + 0.5 * (ah - 1);
    for (int si = 0; si < 3; ++si) {
      const double w2 = aw * scales[si], h2 = ah * scales[si];
      const int k = (ri * 3 + si) * 4;
      out36[k + 0] = (float)(acx - 0.5 * (w2 - 1));
      out36[k + 1] = (float)(acy - 0.5 * (h2 - 1));
      out36[k + 2] = (float)(acx + 0.5 * (w2 - 1));
      out36[k + 3] = (float)(acy + 0.5 * (h2 - 1));
    }
  }
}

extern "C" void kernel_launch(void* const* d_in, const int* in_sizes, int n_in,
                              void* d_out, int out_size, void* d_ws, size_t ws_size,
                              hipStream_t stream)
{
  const float* img      = (const float*)d_in[0];
  const float* info     = (const float*)d_in[1];
  const float* conv1_w  = (const float*)d_in[2];
  const float* conv1_b  = (const float*)d_in[3];
  const float* score_w  = (const float*)d_in[4];
  const float* score_b  = (const float*)d_in[5];
  const float* bbox_w   = (const float*)d_in[6];
  const float* bbox_b   = (const float*)d_in[7];
  const float* fc6_w    = (const float*)d_in[8];
  const float* fc6_b    = (const float*)d_in[9];
  const float* fc7_w    = (const float*)d_in[10];
  const float* fc7_b    = (const float*)d_in[11];
  const float* sfc_w    = (const float*)d_in[12];
  const float* sfc_b    = (const float*)d_in[13];
  const float* bfc_w    = (const float*)d_in[14];
  const float* bfc_b    = (const float*)d_in[15];
  (void)in_sizes; (void)n_in; (void)out_size; (void)ws_size;

  float* ws    = (float*)d_ws;
  float* rpn   = ws + OFF_RPN;
  float* clsS  = ws + OFF_CLS;
  float* bbxP  = ws + OFF_BBX;
  float* props = ws + OFF_PROPS;
  float* score = ws + OFF_SCORE;
  float* scnms = ws + OFF_SCNMS;
  float* rois  = ws + OFF_ROIS;
  float* xp    = ws + OFF_XP;
  float* x2    = ws + OFF_X2;
  float* x3    = ws + OFF_X3;
  float* craw  = ws + OFF_CRAW;

  float* out_cls  = (float*)d_out;                 // 300 x 32
  float* out_box  = out_cls + POST_NMS * NCLS;     // 300 x 128
  float* out_rois = out_box + POST_NMS * 4 * NCLS; // 300 x 4

  AnchorTab at; gen_anchors(at.a);

  // RPN conv (WMMA implicit GEMM)
  rpn_conv_wmma<<<dim3(CIN / 16, (SPAT + 15) / 16), 32, 0, stream>>>(
      img, conv1_w, conv1_b, rpn);

  // 1x1 heads
  rpn_heads<<<(SPAT * 54 + 255) / 256, 256, 0, stream>>>(
      rpn, score_w, score_b, bbox_w, bbox_b, clsS, bbxP);

  // proposals
  proposals_kernel<<<(NTOT + 255) / 256, 256, 0, stream>>>(
      clsS, bbxP, info, at, props, score);

  // top-6000 mask (regenerated every call -> deterministic)
  topk_mask<<<(NTOT + 255) / 256, 256, 0, stream>>>(score, scnms);

  // sequential NMS
  nms_kernel<<<1, 1024, 0, stream>>>(props, scnms, rois, out_rois);

  // zero the 4 pad rows of XP, then ROI pool
  zero_fill<<<((size_t)4 * FCK + 255) / 256, 256, 0, stream>>>(
      xp + (size_t)POST_NMS * FCK, (size_t)4 * FCK);
  roi_pool_kernel<<<(POST_NMS * CIN * POOL * POOL + 255) / 256, 256, 0, stream>>>(
      img, rois, xp);

  // FC stack (WMMA f32; weight-bandwidth-optimal schedule)
  fc_gemm_wmma<MT_FC><<<4096 / 16, 32, 0, stream>>>(xp, fc6_w, fc6_b, x2,
                                                    FCK, 4096, MPAD, 1);
  fc_gemm_wmma<MT_FC><<<4096 / 16, 32, 0, stream>>>(x2, fc7_w, fc7_b, x3,
                                                    4096, 4096, MPAD, 1);
  fc_gemm_wmma<MT_FC><<<NCLS / 16, 32, 0, stream>>>(x3, sfc_w, sfc_b, craw,
                                                    4096, NCLS, MPAD, 0);
  fc_gemm_wmma<MT_FC><<<(4 * NCLS) / 16, 32, 0, stream>>>(x3, bfc_w, bfc_b, out_box,
                                                          4096, 4 * NCLS, POST_NMS, 0);

  // final softmax
  softmax32<<<(POST_NMS + 63) / 64, 64, 0, stream>>>(craw, out_cls);
}